// NodeUnpool_52312701665805
// MI455X (gfx1250) — compile-verified
//
#include <hip/hip_runtime.h>

typedef float v2f __attribute__((ext_vector_type(2)));
typedef float v4f __attribute__((ext_vector_type(4)));
typedef float v8f __attribute__((ext_vector_type(8)));

#define DIM 256
#define WPITCH 260  // LDS row pitch in floats: 260 % 64 = 4 -> conflict-free column reads

// Kernel A: copy h_full -> out. Rows selected by old_idxs are overwritten by
// the GEMM/scatter kernel afterwards (same stream => ordered).
__global__ void copy_full_kernel(const float4* __restrict__ src,
                                 float4* __restrict__ dst, size_t n4) {
  size_t i = (size_t)blockIdx.x * blockDim.x + threadIdx.x;
  size_t stride = (size_t)gridDim.x * blockDim.x;
  for (; i < n4; i += stride) dst[i] = src[i];
}

// Kernel B: merged = h_full[idx] @ W1^T + b1 + h_sub @ W2^T + b2, scattered
// into out. One wave32 per 16-row strip; 8 waves per block share LDS-staged
// weight tiles. Uses V_WMMA_F32_16X16X4_F32 (K-step 4, fp32 exact).
__global__ __launch_bounds__(256) void merge_scatter_kernel(
    const float* __restrict__ h_full, const float* __restrict__ h_sub,
    const float* __restrict__ W1, const float* __restrict__ b1,
    const float* __restrict__ W2, const float* __restrict__ b2,
    const int* __restrict__ idx, float* __restrict__ out, int M) {
  __shared__ float sW[2][16][WPITCH];  // [W1|W2][out-col within tile][K]

  const int lane = threadIdx.x & 31;
  const int waveInBlock = threadIdx.x >> 5;            // 0..7
  const int strip = blockIdx.x * 8 + waveInBlock;      // 16-row strip id
  const int nStrips = M >> 4;                          // M is a multiple of 16
  const bool active = strip < nStrips;                 // uniform per wave
  const int row0 = (active ? strip : 0) << 4;

  // A-operand layout for V_WMMA_F32_16X16X4_F32 (16x4, 2 VGPRs):
  //   lanes 0-15 : M = lane,    supply K = k, k+1
  //   lanes 16-31: M = lane-16, supply K = k+2, k+3
  const int mLocal = lane & 15;
  const int koff = (lane < 16) ? 0 : 2;
  const float* oldRow = h_full + (size_t)idx[row0 + mLocal] * DIM + koff;
  const float* subRow = h_sub + (size_t)(row0 + mLocal) * DIM + koff;

  // C/D layout: VGPR i, lanes 0-15 -> M = i; lanes 16-31 -> M = 8 + i.
  size_t outRow[8];
  {
    const int mBase = (lane < 16) ? 0 : 8;
#pragma unroll
    for (int i = 0; i < 8; ++i)
      outRow[i] = (size_t)idx[row0 + mBase + i] * DIM;
  }

  for (int n0 = 0; n0 < DIM; n0 += 16) {
    // ---- cooperatively stage W1/W2 rows n0..n0+15 (all K) into LDS ----
    __syncthreads();
#pragma unroll
    for (int j = 0; j < 8; ++j) {
      int e = j * 256 + threadIdx.x;  // 2048 float4 elements total
      int mat = e >> 10;              // 0: W1, 1: W2
      int rem = e & 1023;
      int r = rem >> 6;               // weight row = output column 0..15
      int c4 = rem & 63;              // float4 column along K
      const float* src = (mat ? W2 : W1) + (size_t)(n0 + r) * DIM + c4 * 4;
      *(v4f*)&sW[mat][r][c4 * 4] = *(const v4f*)src;
    }
    __syncthreads();

    // B-operand (4x16): B[kk][n] = W[n0+n][k+kk].
    //   lanes 0-15 : N = lane,    rows k, k+1
    //   lanes 16-31: N = lane-16, rows k+2, k+3
    const float* bw1 = &sW[0][mLocal][koff];
    const float* bw2 = &sW[1][mLocal][koff];

    v8f acc = {};
#pragma unroll 8
    for (int k = 0; k < DIM; k += 4) {
      v2f a1 = *(const v2f*)(oldRow + k);
      v2f a2 = *(const v2f*)(subRow + k);
      v2f w1 = *(const v2f*)(bw1 + k);
      v2f w2 = *(const v2f*)(bw2 + k);
      acc = __builtin_amdgcn_wmma_f32_16x16x4_f32(false, a1, false, w1,
                                                  (short)0, acc, false, false);
      acc = __builtin_amdgcn_wmma_f32_16x16x4_f32(false, a2, false, w2,
                                                  (short)0, acc, false, false);
    }

    if (active) {
      const int n = n0 + mLocal;
      const float bias = b1[n] + b2[n];
#pragma unroll
      for (int i = 0; i < 8; ++i)
        out[outRow[i] + n] = acc[i] + bias;
    }
  }
}

extern "C" void kernel_launch(void* const* d_in, const int* in_sizes, int n_in,
                              void* d_out, int out_size, void* d_ws, size_t ws_size,
                              hipStream_t stream) {
  const float* h_full = (const float*)d_in[0];
  const float* h_sub  = (const float*)d_in[1];
  const float* W1     = (const float*)d_in[2];
  const float* b1     = (const float*)d_in[3];
  const float* W2     = (const float*)d_in[4];
  const float* b2     = (const float*)d_in[5];
  const int*   idx    = (const int*)d_in[6];   // JAX default x64-off => int32
  float* out = (float*)d_out;

  const size_t total = (size_t)in_sizes[0];    // N * DIM
  const int M = in_sizes[6];

  // 1) copy-through of the full node table
  size_t n4 = total / 4;
  int copyBlocks = (int)((n4 + 255) / 256);
  copy_full_kernel<<<copyBlocks, 256, 0, stream>>>(
      (const float4*)h_full, (float4*)out, n4);

  // 2) dual-GEMM merge + scatter of the M pooled rows
  int nStrips = M / 16;
  int gemmBlocks = (nStrips + 7) / 8;
  merge_scatter_kernel<<<gemmBlocks, 256, 0, stream>>>(
      h_full, h_sub, W1, b1, W2, b2, idx, out, M);
}